// CrossSpaceAttention_13297218748873
// MI455X (gfx1250) — compile-verified
//
#include <hip/hip_runtime.h>
#include <math.h>

// ---------------------------------------------------------------------------
// CrossSpaceAttention for MI455X (gfx1250, wave32, WMMA).
// fp32 end-to-end; all GEMM-shaped work on V_WMMA_F32_16X16X4_F32.
// pw_conv stages its activation tile via global_load_async_to_lds_b128
// (ASYNCcnt path) to kill 8x duplicated per-wave VMEM reads.
// ---------------------------------------------------------------------------

typedef __attribute__((ext_vector_type(2))) float v2f;
typedef __attribute__((ext_vector_type(8))) float v8f;

#define BATCH   8
#define CIN     128
#define C2      256
#define HH      128
#define WW      128
#define PP      (HH * WW)      // 16384 spatial positions
#define NHEADS  8
#define CHEAD   32             // C2 / NHEADS

// D = A(16x4) * B(4x16) + C(16x16), fp32 exact
#define WMMA_F32(a, b, c) \
    __builtin_amdgcn_wmma_f32_16x16x4_f32(false, (a), false, (b), (short)0, (c), false, false)

__device__ __forceinline__ v2f ld2(const float* p) {
    const float2 t = *(const float2*)p;
    v2f r; r.x = t.x; r.y = t.y; return r;
}

// ---------------------------------------------------------------------------
// Kernel 1: 1x1 conv == GEMM  Y[b,co,p] = sum_ci W[co,ci] * X[b,ci,p] + bias.
// grid (PP/64, C2/128, BATCH), block 256 (8 waves). Wave -> 16(M) x 64(N) tile.
// The 128(K) x 64(N) activation tile is shared by all 8 waves, so it is staged
// once into LDS with async global->LDS B128 copies (ASYNCcnt), then WMMA
// B-operands come from LDS. Row stride 72 floats: 288B rows keep 16B alignment
// for B128 writes and rotate banks by 8/row so the two lane-halves (rows k,
// k+2) hit disjoint banks.
// ---------------------------------------------------------------------------
__global__ __launch_bounds__(256) void pw_conv_kernel(
    const float* __restrict__ X, const float* __restrict__ Wt,
    const float* __restrict__ bias, float* __restrict__ Y)
{
    __shared__ float xt[CIN][72];   // 36,864 B

    const int tid  = threadIdx.x;
    const int wave = tid >> 5;
    const int lane = tid & 31;
    const int row  = lane & 15;
    const int hi   = lane >> 4;
    const int kk   = hi * 2;

    const int p0  = blockIdx.x * 64;
    const int co0 = blockIdx.y * 128 + wave * 16;
    const int b   = blockIdx.z;

    const float* xb = X + (size_t)b * CIN * PP + p0;

    // Cooperative async copy: 128 rows x 64 floats = 2048 B128 chunks, 8/thread.
    #pragma unroll
    for (int i = 0; i < 8; ++i) {
        const int c  = tid + i * 256;        // chunk id
        const int kr = c >> 4;               // K row (0..127)
        const int c4 = (c & 15) * 4;         // float offset in row (0..60)
        const float*   g = xb + (size_t)kr * PP + c4;
        const unsigned l = (unsigned)(uintptr_t)&xt[kr][c4];
        asm volatile("global_load_async_to_lds_b128 %0, %1, off"
                     :: "v"(l), "v"(g) : "memory");
    }
    asm volatile("s_wait_asynccnt 0x0" ::: "memory");
    __syncthreads();

    v8f acc0 = {}, acc1 = {}, acc2 = {}, acc3 = {};
    for (int k0 = 0; k0 < CIN; k0 += 4) {
        v2f a = ld2(&Wt[(size_t)(co0 + row) * CIN + k0 + kk]);  // K contiguous
        const float* l0 = &xt[k0 + kk][0];
        const float* l1 = &xt[k0 + kk + 1][0];
        v2f bv;
        bv.x = l0[row];      bv.y = l1[row];      acc0 = WMMA_F32(a, bv, acc0);
        bv.x = l0[16 + row]; bv.y = l1[16 + row]; acc1 = WMMA_F32(a, bv, acc1);
        bv.x = l0[32 + row]; bv.y = l1[32 + row]; acc2 = WMMA_F32(a, bv, acc2);
        bv.x = l0[48 + row]; bv.y = l1[48 + row]; acc3 = WMMA_F32(a, bv, acc3);
    }
    // C/D layout: VGPR r -> M=r (lanes 0-15), M=r+8 (lanes 16-31); N = lane&15.
    #pragma unroll
    for (int r = 0; r < 8; ++r) {
        const int orow = co0 + r + hi * 8;
        const float bval = bias[orow];
        float* yb = Y + ((size_t)b * C2 + orow) * PP + p0;
        yb[row]      = acc0[r] + bval;
        yb[16 + row] = acc1[r] + bval;
        yb[32 + row] = acc2[r] + bval;
        yb[48 + row] = acc3[r] + bval;
    }
}

// ---------------------------------------------------------------------------
// Kernel 2: depthwise 3x3 (SAME, zero pad) + bias; optional deterministic
// per-row sum-of-squares partials (for L2 norms of q,k).
// grid (HH, C2, BATCH), block WW=128.
// ---------------------------------------------------------------------------
__global__ __launch_bounds__(128) void dw_conv_kernel(
    const float* __restrict__ Xin, const float* __restrict__ Wd,
    const float* __restrict__ bias, float* __restrict__ Yout,
    float* __restrict__ normPart)
{
    __shared__ float red[128];
    const int x  = threadIdx.x;
    const int y  = blockIdx.x;
    const int ch = blockIdx.y;
    const int b  = blockIdx.z;

    const float* src = Xin + ((size_t)b * C2 + ch) * PP;
    const float* w9  = Wd + ch * 9;

    float sum = bias[ch];
    #pragma unroll
    for (int dy = -1; dy <= 1; ++dy) {
        const int yy = y + dy;
        if (yy < 0 || yy >= HH) continue;
        const float* rowp = src + yy * WW;
        #pragma unroll
        for (int dx = -1; dx <= 1; ++dx) {
            const int xx = x + dx;
            if (xx < 0 || xx >= WW) continue;
            sum += w9[(dy + 1) * 3 + (dx + 1)] * rowp[xx];
        }
    }
    Yout[((size_t)b * C2 + ch) * PP + y * WW + x] = sum;

    if (normPart) {  // deterministic two-pass norm: one partial per image row
        red[x] = sum * sum;
        __syncthreads();
        for (int s = 64; s > 0; s >>= 1) {
            if (x < s) red[x] += red[x + s];
            __syncthreads();
        }
        if (x == 0) normPart[((size_t)b * C2 + ch) * HH + y] = red[0];
    }
}

// Reduce 128 row-partials -> one sum-of-squares per (b, channel).
__global__ __launch_bounds__(128) void norm_reduce_kernel(
    const float* __restrict__ part, float* __restrict__ norm2)
{
    __shared__ float red[128];
    const int t = threadIdx.x;
    red[t] = part[(size_t)blockIdx.x * 128 + t];
    __syncthreads();
    for (int s = 64; s > 0; s >>= 1) {
        if (t < s) red[t] += red[t + s];
        __syncthreads();
    }
    if (t == 0) norm2[blockIdx.x] = red[0];
}

// ---------------------------------------------------------------------------
// Kernel 3: per (b,h) Gram S[c,d] = sum_n q[c,n] k[d,n]; normalize by
// max(||q_c||,eps)*max(||k_d||,eps), scale 0.25, row softmax over d.
// grid 64, block 256 (8 waves split K=16384). WMMA f32, LDS cross-wave reduce.
// ---------------------------------------------------------------------------
__global__ __launch_bounds__(256) void attn_gram_kernel(
    const float* __restrict__ Q2, const float* __restrict__ K2,
    const float* __restrict__ qn2, const float* __restrict__ kn2,
    float* __restrict__ Aout)
{
    __shared__ float sred[8 * 1024];   // 8 waves x (4 tiles x 8 regs x 32 lanes)
    __shared__ float Smat[32][33];

    const int tid  = threadIdx.x;
    const int wave = tid >> 5;
    const int lane = tid & 31;
    const int row  = lane & 15;
    const int kk   = (lane >> 4) * 2;

    const int bh  = blockIdx.x;
    const int b   = bh / NHEADS;
    const int h   = bh % NHEADS;
    const int hc0 = h * CHEAD;

    const float* qbase = Q2 + ((size_t)b * C2 + hc0) * PP;
    const float* kbase = K2 + ((size_t)b * C2 + hc0) * PP;

    v8f a00 = {}, a01 = {}, a10 = {}, a11 = {};
    const int n0 = wave * (PP / 8);
    for (int n = n0; n < n0 + PP / 8; n += 4) {
        v2f qa0 = ld2(&qbase[(size_t)row * PP + n + kk]);        // q rows 0..15
        v2f qa1 = ld2(&qbase[(size_t)(16 + row) * PP + n + kk]); // q rows 16..31
        v2f kb0 = ld2(&kbase[(size_t)row * PP + n + kk]);        // k cols 0..15
        v2f kb1 = ld2(&kbase[(size_t)(16 + row) * PP + n + kk]); // k cols 16..31
        a00 = WMMA_F32(qa0, kb0, a00);
        a01 = WMMA_F32(qa0, kb1, a01);
        a10 = WMMA_F32(qa1, kb0, a10);
        a11 = WMMA_F32(qa1, kb1, a11);
    }
    float* wslot = &sred[wave * 1024];
    #pragma unroll
    for (int r = 0; r < 8; ++r) {
        wslot[0 * 256 + r * 32 + lane] = a00[r];
        wslot[1 * 256 + r * 32 + lane] = a01[r];
        wslot[2 * 256 + r * 32 + lane] = a10[r];
        wslot[3 * 256 + r * 32 + lane] = a11[r];
    }
    __syncthreads();

    for (int e = tid; e < 1024; e += 256) {
        float s = 0.f;
        #pragma unroll
        for (int w = 0; w < 8; ++w) s += sred[w * 1024 + e];
        const int tile = e >> 8, r = (e >> 5) & 7, ln = e & 31;
        const int mt = tile >> 1, nt = tile & 1;
        const int c = mt * 16 + r + ((ln >= 16) ? 8 : 0);
        const int d = nt * 16 + (ln & 15);
        Smat[c][d] = s;
    }
    __syncthreads();

    // softmax over d (32 lanes == one row), normalization folded in
    const float scale = 0.25f;  // 1/sqrt(CIN/NHEADS)
    const float kn = fmaxf(sqrtf(kn2[b * C2 + hc0 + lane]), 1e-12f);
    #pragma unroll
    for (int i = 0; i < 4; ++i) {
        const int c = wave * 4 + i;
        const float qn = fmaxf(sqrtf(qn2[b * C2 + hc0 + c]), 1e-12f);
        float val = Smat[c][lane] / (qn * kn) * scale;
        float m = val;
        for (int off = 16; off >= 1; off >>= 1) m = fmaxf(m, __shfl_xor(m, off, 32));
        const float ev = __expf(val - m);
        float sm = ev;
        for (int off = 16; off >= 1; off >>= 1) sm += __shfl_xor(sm, off, 32);
        Aout[((size_t)bh * 32 + c) * 32 + lane] = ev / sm;
    }
}

// ---------------------------------------------------------------------------
// Kernel 4: fused (attn @ v) -> LDS -> output projection. The 134 MB attn@v
// tensor never touches HBM. grid (PP/64, BATCH), block 256 (8 waves).
// LDS att stride 65 floats (bank-conflict free).
// ---------------------------------------------------------------------------
__global__ __launch_bounds__(256) void attn_apply_proj_kernel(
    const float* __restrict__ V2, const float* __restrict__ Amat,
    const float* __restrict__ PWt, const float* __restrict__ PB,
    float* __restrict__ Out)
{
    __shared__ float att[C2][65];
    const int tid  = threadIdx.x;
    const int wave = tid >> 5;
    const int lane = tid & 31;
    const int row  = lane & 15;
    const int hi   = lane >> 4;
    const int kk   = hi * 2;

    const int p0 = blockIdx.x * 64;
    const int b  = blockIdx.y;

    // Stage 1: att[hc][col] = sum_d A[b,h,c,d] * v[b, h*32+d, p0+col]  (K=32)
    for (int s = wave * 2; s < wave * 2 + 2; ++s) {
        const int head = s >> 1;
        const int mt   = s & 1;
        const float* arow  = Amat + ((size_t)(b * NHEADS + head) * 32 + mt * 16 + row) * 32;
        const float* vbase = V2 + ((size_t)b * C2 + head * 32) * PP + p0;
        v8f c0 = {}, c1 = {}, c2 = {}, c3 = {};
        #pragma unroll
        for (int k0 = 0; k0 < 32; k0 += 4) {
            v2f a = ld2(&arow[k0 + kk]);
            const float* v0 = vbase + (size_t)(k0 + kk) * PP;
            const float* v1 = v0 + PP;
            v2f bv;
            bv.x = v0[row];      bv.y = v1[row];      c0 = WMMA_F32(a, bv, c0);
            bv.x = v0[16 + row]; bv.y = v1[16 + row]; c1 = WMMA_F32(a, bv, c1);
            bv.x = v0[32 + row]; bv.y = v1[32 + row]; c2 = WMMA_F32(a, bv, c2);
            bv.x = v0[48 + row]; bv.y = v1[48 + row]; c3 = WMMA_F32(a, bv, c3);
        }
        const int r0 = head * 32 + mt * 16 + hi * 8;
        #pragma unroll
        for (int r = 0; r < 8; ++r) {
            att[r0 + r][row]      = c0[r];
            att[r0 + r][16 + row] = c1[r];
            att[r0 + r][32 + row] = c2[r];
            att[r0 + r][48 + row] = c3[r];
        }
    }
    __syncthreads();

    // Stage 2: Out[b,co,p] = sum_hc PW[co,hc] * att[hc][col] + PB[co]  (K=256)
    {
        const int co0 = wave * 16;
        v8f c0 = {}, c1 = {}, c2 = {}, c3 = {};
        for (int k0 = 0; k0 < C2; k0 += 4) {
            v2f a = ld2(&PWt[(size_t)(co0 + row) * C2 + k0 + kk]);
            const float* l0 = &att[k0 + kk][0];
            const float* l1 = &att[k0 + kk + 1][0];
            v2f bv;
            bv.x = l0[row];      bv.y = l1[row];      c0 = WMMA_F32(a, bv, c0);
            bv.x = l0[16 + row]; bv.y = l1[16 + row]; c1 = WMMA_F32(a, bv, c1);
            bv.x = l0[32 + row]; bv.y = l1[32 + row]; c2 = WMMA_F32(a, bv, c2);
            bv.x = l0[48 + row]; bv.y = l1[48 + row]; c3 = WMMA_F32(a, bv, c3);
        }
        #pragma unroll
        for (int r = 0; r < 8; ++r) {
            const int orow = co0 + r + hi * 8;
            const float bval = PB[orow];
            float* ob = Out + ((size_t)b * CIN + orow) * PP + p0;
            ob[row]      = c0[r] + bval;
            ob[16 + row] = c1[r] + bval;
            ob[32 + row] = c2[r] + bval;
            ob[48 + row] = c3[r] + bval;
        }
    }
}

// ---------------------------------------------------------------------------
extern "C" void kernel_launch(void* const* d_in, const int* in_sizes, int n_in,
                              void* d_out, int out_size, void* d_ws, size_t ws_size,
                              hipStream_t stream) {
    (void)in_sizes; (void)n_in; (void)out_size; (void)ws_size;

    const float* x0  = (const float*)d_in[0];
    const float* x1  = (const float*)d_in[1];
    const float* x2  = (const float*)d_in[2];
    const float* qw  = (const float*)d_in[3];
    const float* qb  = (const float*)d_in[4];
    const float* kw  = (const float*)d_in[5];
    const float* kb  = (const float*)d_in[6];
    const float* vw  = (const float*)d_in[7];
    const float* vb  = (const float*)d_in[8];
    const float* qdw = (const float*)d_in[9];
    const float* qdb = (const float*)d_in[10];
    const float* kdw = (const float*)d_in[11];
    const float* kdb = (const float*)d_in[12];
    const float* vdw = (const float*)d_in[13];
    const float* vdb = (const float*)d_in[14];
    const float* pw  = (const float*)d_in[15];
    const float* pb  = (const float*)d_in[16];
    float* out = (float*)d_out;

    float* ws = (float*)d_ws;
    const size_t N1 = (size_t)BATCH * C2 * PP;   // 33,554,432 floats
    float* tmp  = ws;                 // reused for q1/k1/v1 serially
    float* q2   = ws + N1;
    float* k2   = ws + 2 * N1;
    float* v2   = ws + 3 * N1;
    float* part = ws + 4 * N1;                        // B*C2*H partials
    float* qn2  = part + (size_t)BATCH * C2 * HH;     // B*C2 sum-of-squares
    float* kn2  = qn2 + BATCH * C2;
    float* attn = kn2 + BATCH * C2;                   // B*NHEADS*32*32

    const dim3 gPW(PP / 64, C2 / 128, BATCH);
    const dim3 gDW(HH, C2, BATCH);

    // q path
    pw_conv_kernel<<<gPW, 256, 0, stream>>>(x0, qw, qb, tmp);
    dw_conv_kernel<<<gDW, 128, 0, stream>>>(tmp, qdw, qdb, q2, part);
    norm_reduce_kernel<<<BATCH * C2, 128, 0, stream>>>(part, qn2);
    // k path
    pw_conv_kernel<<<gPW, 256, 0, stream>>>(x1, kw, kb, tmp);
    dw_conv_kernel<<<gDW, 128, 0, stream>>>(tmp, kdw, kdb, k2, part);
    norm_reduce_kernel<<<BATCH * C2, 128, 0, stream>>>(part, kn2);
    // v path (no norm)
    pw_conv_kernel<<<gPW, 256, 0, stream>>>(x2, vw, vb, tmp);
    dw_conv_kernel<<<gDW, 128, 0, stream>>>(tmp, vdw, vdb, v2, nullptr);

    attn_gram_kernel<<<BATCH * NHEADS, 256, 0, stream>>>(q2, k2, qn2, kn2, attn);
    attn_apply_proj_kernel<<<dim3(PP / 64, BATCH), 256, 0, stream>>>(v2, attn, pw, pb, out);
}